// GEMTEGraph2D_11742440587918
// MI455X (gfx1250) — compile-verified
//
#include <hip/hip_runtime.h>
#include <cstdint>
#include <cstddef>

// 2D FDTD (TEz) fused step for MI455X / gfx1250.
// HBM roofline: 9 arrays x 64 MB = 604 MB/step -> ~26 us at 23.3 TB/s.
// VALU roofline forced the round-2 restructure: shared 1/mu tile (computed while
// the Ez async DMA is in flight) + single-divide semi-implicit update
// (A_plus*eps = eps + sig*DT/2), cutting f32 divides from ~5.3 to ~2.3 per cell.
// CDNA5 paths: GLOBAL_LOAD_ASYNC_TO_LDS_B32 (ASYNCcnt) + s_wait_asynccnt,
// non-temporal loads/stores for single-use streams.

#define NXg 4096
#define NYg 4096

constexpr float DTc    = 5e-4f;
constexpr float INV2DX = 500.0f;   // 1/(2*1e-3)
constexpr float INV2DY = 500.0f;   // 1/(2*1e-3)

#define TX  8     // tile rows (i, stride-NY axis)
#define TY  64    // tile cols (j, contiguous axis)
#define NTH (TX * TY)

#define EZS (TY + 4)   // 68: Ez LDS row stride (radius-2 halo)
#define MUS (TY + 2)   // 66: 1/mu LDS row stride (radius-1 halo both dims)
#define HXS (TY + 2)   // 66: hx_new LDS row stride (j-halo 1)
#define HYS (TY)       // 64: hy_new LDS row stride (i-halo 1)

__device__ __forceinline__ void async_ld_b32(uint32_t lds_addr, const void* gaddr) {
  // LDS byte address taken from the first VGPR operand; global addr from VGPR pair. Tracked by ASYNCcnt.
  asm volatile("global_load_async_to_lds_b32 %0, %1, off"
               :: "v"(lds_addr), "v"(gaddr) : "memory");
}

__device__ __forceinline__ void wait_async_all() {
  asm volatile("s_wait_asynccnt 0" ::: "memory");
}

__device__ __forceinline__ int clampi(int v, int lo, int hi) {
  return v < lo ? lo : (v > hi ? hi : v);
}

__global__ __launch_bounds__(NTH)
void fdtd_step_kernel(const float* __restrict__ Ez,  const float* __restrict__ Hx,
                      const float* __restrict__ Hy,  const float* __restrict__ eps,
                      const float* __restrict__ mu,  const float* __restrict__ sig,
                      float* __restrict__ ez_out, float* __restrict__ hx_out,
                      float* __restrict__ hy_out)
{
  __shared__ float s_ez [(TX + 4) * EZS];  // Ez with radius-2 halo
  __shared__ float s_rmu[(TX + 2) * MUS];  // 1/mu with radius-1 halo
  __shared__ float s_hxn[ TX      * HXS];  // hx_new, j extended by +-1
  __shared__ float s_hyn[(TX + 2) * HYS];  // hy_new, i extended by +-1

  const int tj  = threadIdx.x;            // 0..63 (contiguous j)
  const int ti  = threadIdx.y;            // 0..7
  const int tid = ti * TY + tj;           // 0..511
  const int j0  = blockIdx.x * TY;
  const int i0  = blockIdx.y * TX;

  // ---------- stage 0: async-stage Ez (TX+4) x (TY+4) halo tile into LDS ----------
  // Out-of-grid halo addresses are clamped; those values are dead (interior masks
  // below zero every term that would consume them), so clamping is safe.
  const uint32_t ez_lds_base = (uint32_t)(uintptr_t)(&s_ez[0]);
  for (int t = tid; t < (TX + 4) * EZS; t += NTH) {
    const int r  = t / EZS;
    const int c  = t - r * EZS;
    const int gi = clampi(i0 - 2 + r, 0, NXg - 1);
    const int gj = clampi(j0 - 2 + c, 0, NYg - 1);
    async_ld_b32(ez_lds_base + 4u * (uint32_t)t, Ez + ((size_t)gi * NYg + gj));
  }

  // ---------- stage 0b (overlaps the async DMA): shared 1/mu halo tile ----------
  // (TX+2) x (TY+2) reciprocals, reused by both hx_new and hy_new stages.
  for (int t = tid; t < (TX + 2) * MUS; t += NTH) {
    const int r  = t / MUS;
    const int c  = t - r * MUS;
    const int gi = clampi(i0 - 1 + r, 0, NXg - 1);
    const int gj = clampi(j0 - 1 + c, 0, NYg - 1);
    s_rmu[t] = 1.0f / mu[(size_t)gi * NYg + gj];
  }

  wait_async_all();       // this wave's async transfers into LDS complete
  __syncthreads();        // every wave has waited -> s_ez and s_rmu fully visible

  // ---------- stage 1a: hx_new on TX x (TY+2)   (needs ez(i, j+-1)) ----------
  for (int t = tid; t < TX * HXS; t += NTH) {
    const int r   = t / HXS;
    const int c   = t - r * HXS;
    const int gi  = i0 + r;                       // always in [0, NXg)
    const int gj  = j0 - 1 + c;                   // may be -1 or NYg at grid edge
    const int gjc = clampi(gj, 0, NYg - 1);
    const bool interior = (gi >= 1) & (gi <= NXg - 2) & (gj >= 1) & (gj <= NYg - 2);
    const float ezp = s_ez[(r + 2) * EZS + (c + 2)];   // ez(gi, gj+1)
    const float ezm = s_ez[(r + 2) * EZS +  c     ];   // ez(gi, gj-1)
    const float dez_dy = interior ? (ezp - ezm) * INV2DY : 0.0f;
    const float rmu = s_rmu[(r + 1) * MUS + c];        // 1/mu(gi, gjc)
    s_hxn[t] = Hx[(size_t)gi * NYg + gjc] - DTc * (dez_dy * rmu);
  }

  // ---------- stage 1b: hy_new on (TX+2) x TY   (needs ez(i+-1, j)) ----------
  for (int t = tid; t < (TX + 2) * HYS; t += NTH) {
    const int r   = t >> 6;                 // / HYS (=64)
    const int c   = t & (HYS - 1);
    const int gi  = i0 - 1 + r;             // may be -1 or NXg at grid edge
    const int gj  = j0 + c;                 // always in [0, NYg)
    const int gic = clampi(gi, 0, NXg - 1);
    const bool interior = (gi >= 1) & (gi <= NXg - 2) & (gj >= 1) & (gj <= NYg - 2);
    const float ezp = s_ez[(r + 2) * EZS + (c + 2)];   // ez(gi+1, gj)
    const float ezm = s_ez[ r      * EZS + (c + 2)];   // ez(gi-1, gj)
    const float dez_dx = interior ? (ezp - ezm) * INV2DX : 0.0f;
    const float rmu = s_rmu[r * MUS + (c + 1)];        // 1/mu(gic, gj)
    s_hyn[t] = Hy[(size_t)gic * NYg + gj] + DTc * (dez_dx * rmu);
  }
  __syncthreads();

  // ---------- stage 2: curl of new H, lossy semi-implicit Ez update ----------
  const int gi = i0 + ti;
  const int gj = j0 + tj;
  const size_t g = (size_t)gi * NYg + gj;

  const float hxc = s_hxn[ti * HXS + (tj + 1)];        // hx_new(gi, gj)
  const float hyc = s_hyn[(ti + 1) * HYS + tj];        // hy_new(gi, gj)

  const bool interior = (gi >= 1) & (gi <= NXg - 2) & (gj >= 1) & (gj <= NYg - 2);
  float curl = 0.0f;
  if (interior) {
    const float dhy_dx = (s_hyn[(ti + 2) * HYS + tj] - s_hyn[ti * HYS + tj]) * INV2DX;
    const float dhx_dy = (s_hxn[ti * HXS + (tj + 2)] - s_hxn[ti * HXS + tj]) * INV2DY;
    curl = dhy_dx - dhx_dy;
  }

  const float ezc = s_ez[(ti + 2) * EZS + (tj + 2)];
  const float e   = __builtin_nontemporal_load(&eps[g]);  // read-once streams
  const float sg  = __builtin_nontemporal_load(&sig[g]);

  // A_plus*eps = eps + sig*DT/2 ; A_minus*eps = eps - sig*DT/2  ->  one divide:
  // ez_new = (A_minus/A_plus)*ez + (DT/(A_plus*eps))*curl
  //        = r*(A_minus*eps)*ez + r*DT*curl, with r = 1/(A_plus*eps)
  const float half_sdt = 0.5f * DTc * sg;
  const float denom    = e + half_sdt;
  const float numer    = e - half_sdt;
  const float r        = 1.0f / denom;
  const float ezn      = (numer * r) * ezc + (DTc * r) * curl;

  __builtin_nontemporal_store(ezn, &ez_out[g]);   // outputs are never re-read
  __builtin_nontemporal_store(hxc, &hx_out[g]);
  __builtin_nontemporal_store(hyc, &hy_out[g]);
}

extern "C" void kernel_launch(void* const* d_in, const int* in_sizes, int n_in,
                              void* d_out, int out_size, void* d_ws, size_t ws_size,
                              hipStream_t stream) {
  (void)in_sizes; (void)n_in; (void)out_size; (void)d_ws; (void)ws_size;
  const float* Ez  = (const float*)d_in[0];
  const float* Hx  = (const float*)d_in[1];
  const float* Hy  = (const float*)d_in[2];
  const float* eps = (const float*)d_in[3];
  const float* mu  = (const float*)d_in[4];
  const float* sig = (const float*)d_in[5];
  float* out = (float*)d_out;
  const size_t N = (size_t)NXg * NYg;

  dim3 block(TY, TX);                 // 64 x 8 = 512 threads = 16 wave32
  dim3 grid(NYg / TY, NXg / TX);      // 64 x 512 blocks, exact tiling of 4096x4096
  fdtd_step_kernel<<<grid, block, 0, stream>>>(Ez, Hx, Hy, eps, mu, sig,
                                               out, out + N, out + 2 * N);
}